// MCPBRNN_Generic_PETconstraint_Three_VariantOutputGate_BYPASSM1_10196252360855
// MI455X (gfx1250) — compile-verified
//
#include <hip/hip_runtime.h>
#include <math.h>

// ---------------------------------------------------------------------------
// MCPBRNN (H=1) on MI455X / gfx1250.
//
//  k1: single-lane sequential recurrence (40000 steps, latency-bound;
//      chain depth ~= one v_exp_f32 + v_rcp_f32 per step since the 5
//      sigmoids are independent given the carry). Stores are posted
//      (STOREcnt) and do not extend the chain. global_prefetch_b8 streams x.
//  k2: obs-std over y_obs[365:10000) via V_WMMA_F32_16X16X4_F32 used as an
//      exact f32 reduction engine (A = data, B = all-ones 4x16). Loads use
//      clamped indices so they are unconditional (no EXEC branching); the
//      29-element tail is zeroed with v_cndmask selects instead.
//  k3: wide fill of h_nout (B,2) and obs_std (B,1).
// ---------------------------------------------------------------------------

typedef __attribute__((ext_vector_type(2))) float v2f;
typedef __attribute__((ext_vector_type(8))) float v8f;

#define SPIN 365
#define TRAIN 10000
#define ML_C 2.9086f
#define SL_C 1.898f
#define U1MAX_C 221.519f

__device__ __forceinline__ float fsig(float v) {
  return 1.0f / (1.0f + __expf(-v));
}

// ---- k1: sequential scalar recurrence, one lane ---------------------------
__global__ void __launch_bounds__(32)
mcpbrnn_recurrence_kernel(
    const float* __restrict__ x,
    const float* __restrict__ p_mean, const float* __restrict__ p_std,
    const float* __restrict__ w_r_yom, const float* __restrict__ w_r_yom_fp,
    const float* __restrict__ w_r_yom_gw, const float* __restrict__ w_r_ylm,
    const float* __restrict__ w_r_yfm,
    const float* __restrict__ b0_yom,  const float* __restrict__ w1_yom,
    const float* __restrict__ b0_gw,   const float* __restrict__ w1_gw,
    const float* __restrict__ b0_fp,   const float* __restrict__ w1_fp,
    const float* __restrict__ b0_ylm,  const float* __restrict__ w2_ylm,
    const float* __restrict__ w1_yum,  const float* __restrict__ b0_yum,
    float* __restrict__ out, int B)
{
  if (threadIdx.x != 0 || blockIdx.x != 0) return;

  const float mo = p_mean[0];
  const float inv_so = 1.0f / p_std[0];

  // softmax-style output-gate scale factors (loop invariants)
  const float e_oo = __expf(w_r_yom[0]);
  const float e_gw = __expf(w_r_yom_gw[0]);
  const float e_l  = __expf(w_r_ylm[0]);
  const float e_f  = __expf(w_r_yfm[0]);
  const float e_fp = __expf(w_r_yom_fp[0]);
  const float inv_den = 1.0f / (e_oo + e_gw + e_l + e_f + e_fp);
  const float oo1   = e_oo * inv_den;
  const float oogw1 = e_gw * inv_den;
  const float oofp1 = e_fp * inv_den;
  const float ol1   = e_l  * inv_den;

  const float cb_yom = b0_yom[0], cw_yom = w1_yom[0];
  const float cb_gw  = b0_gw[0],  cw_gw  = w1_gw[0];
  const float cb_fp  = b0_fp[0],  cw_fp  = w1_fp[0];
  const float cb_ylm = b0_ylm[0], cw_ylm = w2_ylm[0];
  const float cb_yum = b0_yum[0], cw_yum = w1_yum[0];

  const float invU1 = 1.0f / U1MAX_C;
  const float invSL = 1.0f / SL_C;

  float* __restrict__ o_h    = out + 0  * (size_t)B;
  float* __restrict__ o_hfp  = out + 1  * (size_t)B;
  float* __restrict__ o_c    = out + 2  * (size_t)B;
  float* __restrict__ o_l    = out + 3  * (size_t)B;
  float* __restrict__ o_lc   = out + 4  * (size_t)B;
  float* __restrict__ o_bp   = out + 5  * (size_t)B;
  float* __restrict__ o_gw   = out + 6  * (size_t)B;
  float* __restrict__ o_ib   = out + 7  * (size_t)B;
  float* __restrict__ o_goo  = out + 8  * (size_t)B;
  float* __restrict__ o_gfp  = out + 9  * (size_t)B;
  float* __restrict__ o_gol  = out + 10 * (size_t)B;
  float* __restrict__ o_golc = out + 11 * (size_t)B;
  float* __restrict__ o_gf   = out + 12 * (size_t)B;
  float* __restrict__ o_ggw  = out + 13 * (size_t)B;

  float c = 0.0f;
  for (int t = 0; t < B; ++t) {
    const float u1 = x[2 * t];
    const float u2 = x[2 * t + 1];
    __builtin_prefetch(x + 2 * t + 256, 0, 0);  // global_prefetch_b8, ~1KB ahead

    const float zn = (c - mo) * inv_so;

    // 5 independent sigmoids -> chain depth is one exp+rcp, not five
    const float ib   = fsig(cb_yum + zn * cw_yum + (u1 * invU1) * cw_yum);
    const float oo   = oo1   * fsig(cb_yom + zn * cw_yom);
    const float oogw = oogw1 * fsig(cb_gw  + zn * cw_gw);
    const float oofp = oofp1 * fsig(cb_fp  + zn * cw_fp);
    const float ol   = ol1   * fsig(cb_ylm + (u2 - ML_C) * invSL * cw_ylm);

    const bool  pos    = c > 0.0f;
    const float c_safe = pos ? c : 1.0f;
    const float r      = ol - u2 / c_safe;
    const float relu_r = r > 0.0f ? r : 0.0f;
    const float olc    = pos ? (ol - relu_r) : ol;

    const float f  = 1.0f - oo - oofp - oogw - olc;
    const float g  = u1;
    const float bp = ib * g;

    o_h[t]    = oo * c + bp;
    o_hfp[t]  = oofp * c;
    o_c[t]    = c;           // carry at step entry
    o_l[t]    = ol * c;
    o_lc[t]   = olc * c;
    o_bp[t]   = bp;
    o_gw[t]   = oogw * c;
    o_ib[t]   = ib;
    o_goo[t]  = oo;
    o_gfp[t]  = oofp;
    o_gol[t]  = ol;
    o_golc[t] = olc;
    o_gf[t]   = f;
    o_ggw[t]  = oogw;

    c = f * c + (1.0f - ib) * g;
  }
}

// ---- k2: std(y_obs[SPIN:TRAIN], ddof=1) via WMMA f32 reduction ------------
__global__ void __launch_bounds__(32)
obsstd_wmma_kernel(const float* __restrict__ y_obs, float* __restrict__ ws_std)
{
  __shared__ float red[32];
  const int lane  = threadIdx.x;          // single wave32
  const int n     = TRAIN - SPIN;         // 9635 elements
  const int iters = (n + 63) / 64;        // 64 elements per WMMA (A is 16x4)
  const v2f ones  = {1.0f, 1.0f};         // B = 4x16 all-ones (layout-free)

  // pass 1: exact f32 sum. Clamped indices -> unconditional loads (no EXEC
  // branching); tail elements are zeroed with selects.
  v8f acc = {};
  for (int it = 0; it < iters; ++it) {
    const int idx = SPIN + it * 64 + lane * 2;
    const int i0  = idx     < (TRAIN - 1) ? idx     : (TRAIN - 1);
    const int i1  = idx + 1 < (TRAIN - 1) ? idx + 1 : (TRAIN - 1);
    const float v0 = y_obs[i0];           // always in-bounds
    const float v1 = y_obs[i1];
    v2f a;
    a.x = (idx     < TRAIN) ? v0 : 0.0f;  // v_cndmask
    a.y = (idx + 1 < TRAIN) ? v1 : 0.0f;
    acc = __builtin_amdgcn_wmma_f32_16x16x4_f32(
        false, a, false, ones, (short)0, acc, false, false);
  }
  float p = acc[0] + acc[1] + acc[2] + acc[3] + acc[4] + acc[5] + acc[6] + acc[7];
  red[lane] = p;
  __syncthreads();
  // Sum of ALL 256 D entries == 16 * grand total (each column replicates the
  // row sums), independent of the exact C/D lane layout. /16 is exact.
  float tot = 0.0f;
  for (int i = 0; i < 32; ++i) tot += red[i];
  const float mean = (tot * 0.0625f) / (float)n;
  __syncthreads();

  // pass 2: sum of squared deviations (tail contributes exact zeros)
  v8f acc2 = {};
  for (int it = 0; it < iters; ++it) {
    const int idx = SPIN + it * 64 + lane * 2;
    const int i0  = idx     < (TRAIN - 1) ? idx     : (TRAIN - 1);
    const int i1  = idx + 1 < (TRAIN - 1) ? idx + 1 : (TRAIN - 1);
    const float v0 = y_obs[i0];
    const float v1 = y_obs[i1];
    const float d0 = (idx     < TRAIN) ? (v0 - mean) : 0.0f;
    const float d1 = (idx + 1 < TRAIN) ? (v1 - mean) : 0.0f;
    v2f a = {d0 * d0, d1 * d1};
    acc2 = __builtin_amdgcn_wmma_f32_16x16x4_f32(
        false, a, false, ones, (short)0, acc2, false, false);
  }
  float p2 = acc2[0] + acc2[1] + acc2[2] + acc2[3] + acc2[4] + acc2[5] + acc2[6] + acc2[7];
  red[lane] = p2;
  __syncthreads();
  if (lane == 0) {
    float tot2 = 0.0f;
    for (int i = 0; i < 32; ++i) tot2 += red[i];
    ws_std[0] = sqrtf((tot2 * 0.0625f) / (float)(n - 1));
  }
}

// ---- k3: h_nout = concat([h_n, obs_std], axis=1); obs_std fill ------------
__global__ void finalize_kernel(const float* __restrict__ ws_std,
                                float* __restrict__ out, int B)
{
  const int i = blockIdx.x * blockDim.x + threadIdx.x;
  if (i >= B) return;
  const float s = ws_std[0];
  float* __restrict__ h_nout  = out + 14 * (size_t)B;  // (B,2) row-major
  float* __restrict__ obs_out = out + 16 * (size_t)B;  // (B,1)
  h_nout[2 * i]     = out[i];  // h_n
  h_nout[2 * i + 1] = s;
  obs_out[i]        = s;
}

extern "C" void kernel_launch(void* const* d_in, const int* in_sizes, int n_in,
                              void* d_out, int out_size, void* d_ws, size_t ws_size,
                              hipStream_t stream) {
  const float* x     = (const float*)d_in[0];
  // d_in[1]=epoch, d_in[2]=time_lag : unused by the reference math
  const float* y_obs = (const float*)d_in[3];
  const int B = in_sizes[3];          // 40000
  float* out = (float*)d_out;
  float* ws_std = (float*)d_ws;

  mcpbrnn_recurrence_kernel<<<1, 32, 0, stream>>>(
      x,
      (const float*)d_in[4],  (const float*)d_in[5],
      (const float*)d_in[6],  (const float*)d_in[7],
      (const float*)d_in[8],  (const float*)d_in[9],
      (const float*)d_in[10],
      (const float*)d_in[11], (const float*)d_in[12],
      (const float*)d_in[13], (const float*)d_in[14],
      (const float*)d_in[15], (const float*)d_in[16],
      (const float*)d_in[17], (const float*)d_in[18],
      (const float*)d_in[19], (const float*)d_in[20],
      out, B);

  obsstd_wmma_kernel<<<1, 32, 0, stream>>>(y_obs, ws_std);

  finalize_kernel<<<(B + 255) / 256, 256, 0, stream>>>(ws_std, out, B);
}